// SVSAlgorithm_88587995447540
// MI455X (gfx1250) — compile-verified
//
#include <hip/hip_runtime.h>
#include <hip/hip_bf16.h>
#include <stdint.h>

#define T_FRAMES 2048
#define H_DIM 128
#define W_DIM 160
#define TW 16               // interior tile width
#define TH 8                // interior tile height
#define HTW (TW + 2)        // halo tile width  = 18
#define HTH (TH + 2)        // halo tile height = 10
#define NT  (HTW * HTH)     // threads per block = 180
#define HW  (H_DIM * W_DIM)
#define FB  16              // frames per batch (barrier amortization)
#define NB  (T_FRAMES / FB) // 128 batches

// ---- CDNA5 async copy (global -> LDS), tracked by ASYNCcnt ----------------
#define ASYNC_LOAD_F32(ldsoff, voff, sbase)                                   \
  asm volatile("global_load_async_to_lds_b32 %0, %1, %2"                      \
               :: "v"(ldsoff), "v"(voff), "s"(sbase) : "memory")

#if __has_builtin(__builtin_amdgcn_s_wait_asynccnt)
#define WAIT_ASYNC(n) __builtin_amdgcn_s_wait_asynccnt(n)
#else
#define WAIT_ASYNC(n) asm volatile("s_wait_asynccnt %0" :: "n"(n) : "memory")
#endif

// ---- sigmoid via hardware tanh: sigmoid(z) = 0.5 + 0.5*tanh(z/2) ----------
static __device__ __forceinline__ float hw_tanh(float x) {
#if __has_builtin(__builtin_amdgcn_tanhf)
  return __builtin_amdgcn_tanhf(x);
#elif __has_builtin(__builtin_amdgcn_tanh_f32)
  return __builtin_amdgcn_tanh_f32(x);
#else
  float r;
  asm("v_tanh_f32 %0, %1" : "=v"(r) : "v"(x));
  return r;
#endif
}

__global__ __launch_bounds__(NT) void svs_scan_erode_kernel(
    const float* __restrict__ x,       // (T,1,H,W)
    const float* __restrict__ params,  // [d_close, d_open, d_hot]
    const float* __restrict__ HT0,     // (H,W)
    const float* __restrict__ LT0,     // (H,W)
    const float* __restrict__ kern,    // (1,1,3,3), constant value
    float* __restrict__ out)           // (T,1,H,W)
{
  __shared__ float simg[3][FB][NT];  // triple-buffered image batches (2 ahead)
  __shared__ float shot[2][FB][NT];  // double-buffered hot batches

  const int tid = threadIdx.x;           // 0..179
  const int lx  = tid % HTW;             // 0..17 (halo coords)
  const int ly  = tid / HTW;             // 0..9
  const int gx  = (int)blockIdx.x * TW + lx - 1;  // image w
  const int gy  = (int)blockIdx.y * TH + ly - 1;  // image h
  const bool inImg = (gx >= 0) & (gx < W_DIM) & (gy >= 0) & (gy < H_DIM);
  const int cx = gx < 0 ? 0 : (gx >= W_DIM ? W_DIM - 1 : gx);
  const int cy = gy < 0 ? 0 : (gy >= H_DIM ? H_DIM - 1 : gy);
  const int cpix = cy * W_DIM + cx;      // clamped pixel (safe address)

  const float d_close = params[0];
  const float d_open  = params[1];
  const float d_hot   = params[2];
  const float k       = kern[0];         // constant 2/9 kernel
  const float ndh250  = -(d_hot * 250.0f);
  const float hotmask = inImg ? 1.0f : 0.0f;  // out-of-image halo -> hot = 0

  float HT = HT0[cpix];
  float LT = LT0[cpix];

  const bool interior = (lx >= 1) & (lx <= TW) & (ly >= 1) & (ly <= TH);
  // #valid neighbors under SAME zero-padding (image borders)
  const int cntx = ((gx == 0) | (gx == W_DIM - 1)) ? 2 : 3;
  const int cnty = ((gy == 0) | (gy == H_DIM - 1)) ? 2 : 3;
  const float cnt = (float)(cntx * cnty);

  const uint64_t xbase  = (uint64_t)(uintptr_t)x;
  const uint32_t stride = (uint32_t)(HW * 4);        // bytes per frame
  const uint32_t voff0  = (uint32_t)(cpix * 4);      // byte offset in frame 0

  const uint32_t lds_img = (uint32_t)(uintptr_t)&simg[0][0][tid];
  const uint32_t slotsz  = (uint32_t)(NT * 4);       // bytes per frame slot
  const uint32_t bufsz   = (uint32_t)(FB * NT * 4);  // bytes per img buffer

  // prologue: prefetch batches 0 and 1 (2 * FB frames, buffers 0 and 1)
#pragma unroll
  for (int p = 0; p < 2; ++p) {
    uint32_t vo  = voff0 + (uint32_t)p * (uint32_t)FB * stride;
    uint32_t dst = lds_img + (uint32_t)p * bufsz;
#pragma unroll
    for (int j = 0; j < FB; ++j) {
      ASYNC_LOAD_F32(dst, vo, xbase);
      vo  += stride;
      dst += slotsz;
    }
  }

  float* outp = out + (size_t)gy * W_DIM + gx;

  int buf = 0;  // img buffer holding batch b (cycles 0,1,2)
  for (int b = 0; b < NB; ++b) {
    // prefetch batch b+2 into the free buffer, keeping 2 batches in flight
    if (b + 2 < NB) {
      const int nbuf = (buf + 2 >= 3) ? (buf - 1) : (buf + 2);
      uint32_t vo  = voff0 + (uint32_t)(b + 2) * (uint32_t)FB * stride;
      uint32_t dst = lds_img + (uint32_t)nbuf * bufsz;
#pragma unroll
      for (int j = 0; j < FB; ++j) {
        ASYNC_LOAD_F32(dst, vo, xbase);
        vo  += stride;
        dst += slotsz;
      }
      WAIT_ASYNC(2 * FB);   // oldest FB loads (batch b) done, in-order
    } else if (b + 1 < NB) {
      WAIT_ASYNC(FB);
    } else {
      WAIT_ASYNC(0);
    }

    const int half = b & 1;

    // ---- scan phase: FB thread-local recurrence steps, no sync needed ----
    const float* sb = &simg[buf][0][tid];
    float*       hb = &shot[half][0][tid];
#pragma unroll
    for (int j = 0; j < FB; ++j) {
      const float img = sb[j * NT];
      // hot scores against pre-update thresholds
      const float tH = hw_tanh(fmaf(img - HT, 250.0f, ndh250));
      const float tL = hw_tanh(fmaf(LT - img, 250.0f, ndh250));
      const float hot = fmaf(tH + tL, 0.5f, 1.0f) * hotmask;
      HT += (img > HT) ? d_open : -d_close;
      LT += (LT > img) ? -d_open : d_close;
      hb[j * NT] = hot;
    }

    __syncthreads();   // one barrier per FB frames

    // ---- stencil phase: 3x3 box sum per frame of the batch ----
    if (interior) {
      const float* hc = &shot[half][0][tid];
      float* ob = outp + (size_t)(b * FB) * HW;
#pragma unroll
      for (int j = 0; j < FB; ++j) {
        const float* r = hc + j * NT;
        const float S =
            r[-HTW - 1] + r[-HTW] + r[-HTW + 1] +
            r[-1]       + r[0]    + r[1]        +
            r[HTW - 1]  + r[HTW]  + r[HTW + 1];
        const float o = fmaf(-k, cnt - S, 1.0f);   // 1 - k*(cnt - S)
        __builtin_nontemporal_store(fmaxf(o, 0.0f), ob);
        ob += HW;
      }
    }

    buf = (buf + 1 >= 3) ? 0 : (buf + 1);
  }
}

extern "C" void kernel_launch(void* const* d_in, const int* in_sizes, int n_in,
                              void* d_out, int out_size, void* d_ws, size_t ws_size,
                              hipStream_t stream) {
  (void)in_sizes; (void)n_in; (void)out_size; (void)d_ws; (void)ws_size;
  const float* x      = (const float*)d_in[0];
  const float* params = (const float*)d_in[1];
  const float* HT0    = (const float*)d_in[2];
  const float* LT0    = (const float*)d_in[3];
  const float* kern   = (const float*)d_in[4];
  float* out = (float*)d_out;

  dim3 grid(W_DIM / TW, H_DIM / TH);   // 10 x 16 = 160 blocks
  svs_scan_erode_kernel<<<grid, NT, 0, stream>>>(x, params, HT0, LT0, kern, out);
}